// Simulation_87574383165691
// MI455X (gfx1250) — compile-verified
//
#include <hip/hip_runtime.h>

// ---------------------------------------------------------------------------
// Photophysics Monte-Carlo, parallelized by associative map composition +
// WMMA-f32 triangular-matrix prefix scan (CDNA5 / gfx1250, wave32).
//
// Reference:
//   state chain (4 states), photon = (t==1) which only fires from S0,
//   dwell = -log1p(-ue)/rate[state], cum_time = cumsum(dwell, axis=steps),
//   ind = searchsorted(arange(200000)*0.002, cum_time, 'left'),
//   hist = bincount(ind * photon, minlength=200001).
//
// d_out layout (float32): [photons 20000*2048][cum_time 20000*2048][hist 200001]
// ---------------------------------------------------------------------------

#define STEPS   20000
#define NS      2048
#define CHUNK   80            // steps per chunk (5 groups of 16)
#define NCHUNK  250           // 250*80 = 20000
#define NBINS   200000
#define WAVES   4             // waves per block in the WMMA pass

typedef float v2f __attribute__((ext_vector_type(2)));
typedef float v8f __attribute__((ext_vector_type(8)));

// rate sums per state: S0=1e6, S1=1.2302e8, S2=1e5, S3=0.2
__device__ __forceinline__ float invrate(int s) {
  return s == 0 ? 1.0e-6f
       : s == 1 ? (float)(1.0 / 123020000.0)
       : s == 2 ? 1.0e-5f
       : 5.0f;
}

// transition: S0 -> S1 (photon); S1 -> {0,2,3,0}[G]; S2,S3 -> S0
__device__ __forceinline__ int trans(int s, int G) {
  return (s == 0) ? 1 : (s == 1) ? ((G == 1) ? 2 : (G == 2) ? 3 : 0) : 0;
}

// searchsorted(arange(NBINS)*0.002f, ct, side='left') with float-edge fixups
__device__ __forceinline__ int binIndex(float ct) {
  int i = (int)ceilf(ct * 500.0f);
  if (i < 0) i = 0;
  if (i > NBINS) i = NBINS;
  if (i > 0 && (float)(i - 1) * 0.002f >= ct) --i;
  else if (i < NBINS && (float)i * 0.002f < ct) ++i;
  return i;
}

// ---------------------------------------------------------------------------
// Pass 1: per (emitter, chunk): pack 2-bit S1-branch codes, compose the
// chunk's 4-state transition map, and accumulate dwell sums for all 4
// possible entry states (so no later pass ever re-reads u_cat).
// ---------------------------------------------------------------------------
__global__ __launch_bounds__(256)
void k1_codes_maps_sums(const float* __restrict__ u_cat,
                        const float* __restrict__ u_exp,
                        unsigned int* __restrict__ gcodes,   // [(c*5+w)*NS + e]
                        unsigned int* __restrict__ maps,     // [c*NS + e]
                        float* __restrict__ sums4) {         // [(c*NS+e)*4 + k]
  const int e = blockIdx.x * blockDim.x + threadIdx.x;   // 0..2047
  const int c = blockIdx.y;                              // 0..249
  const float C0 = (float)(50000000.0 / 123020000.0);    // cdf S1 -> t=0
  const float C2 = (float)(53000000.0 / 123020000.0);    // cdf S1 -> t=2
  const float C3 = (float)(53020000.0 / 123020000.0);    // cdf S1 -> t=3

  int s0 = 0, s1 = 1, s2 = 2, s3 = 3;                    // 4 entry-state chains
  float a0 = 0.f, a1 = 0.f, a2 = 0.f, a3 = 0.f;
  unsigned int gw = 0;
  const long base = (long)c * CHUNK * NS + e;

  for (int i = 0; i < CHUNK; ++i) {
    if ((i & 15) == 0)   // stream-ahead hint -> global_prefetch_b8
      __builtin_prefetch(&u_exp[base + (long)(i + 32) * NS], 0, 1);
    const float uc = u_cat[base + (long)i * NS];
    const float ue = u_exp[base + (long)i * NS];
    const int G = uc < C0 ? 0 : uc < C2 ? 1 : uc < C3 ? 2 : 3;
    const float L = -log1pf(-ue);
    a0 += L * invrate(s0);  a1 += L * invrate(s1);
    a2 += L * invrate(s2);  a3 += L * invrate(s3);
    s0 = trans(s0, G); s1 = trans(s1, G); s2 = trans(s2, G); s3 = trans(s3, G);
    gw |= (unsigned int)G << (2 * (i & 15));
    if ((i & 15) == 15) { gcodes[((long)c * 5 + (i >> 4)) * NS + e] = gw; gw = 0; }
  }
  maps[(long)c * NS + e] = (unsigned int)s0 | ((unsigned int)s1 << 8) |
                           ((unsigned int)s2 << 16) | ((unsigned int)s3 << 24);
  float* sp = sums4 + ((long)c * NS + e) * 4;
  sp[0] = a0; sp[1] = a1; sp[2] = a2; sp[3] = a3;
}

// ---------------------------------------------------------------------------
// Pass 2: tiny per-emitter scan over 250 chunks: entry state + time offset.
// ---------------------------------------------------------------------------
__global__ __launch_bounds__(256)
void k2_scan(const unsigned int* __restrict__ maps,
             const float* __restrict__ sums4,
             unsigned char* __restrict__ entry,
             float* __restrict__ off) {
  const int e = blockIdx.x * blockDim.x + threadIdx.x;
  int st = 0; float run = 0.f;
  for (int c = 0; c < NCHUNK; ++c) {
    const long b = (long)c * NS + e;
    entry[b] = (unsigned char)st;
    off[b]   = run;
    run += sums4[b * 4 + st];
    st = (int)((maps[b] >> (8 * st)) & 3u);
  }
}

// ---------------------------------------------------------------------------
// Pass 3: replay with known entry state; WMMA-f32 prefix scan for cum_time.
// Wave owns 32 emitters. Per 16-step group: stage dwell tile (16x32) in LDS,
// then for each 16-emitter half compute D = sum_q Aq(16x4)*Bq(4x16) + C where
// Aq are lower-triangular-ones slabs and C broadcasts the running carry.
// D row 15 (VGPR7, lanes 16..31) is the new carry, fetched via ds_bpermute.
// ---------------------------------------------------------------------------
__global__ __launch_bounds__(WAVES * 32)
void k3_finalize(const float* __restrict__ u_exp,
                 const unsigned int* __restrict__ gcodes,
                 const unsigned char* __restrict__ entry,
                 const float* __restrict__ off,
                 float* __restrict__ out_ph,
                 float* __restrict__ out_ct,
                 int* __restrict__ hist) {
  __shared__ float tile[WAVES][16][33];   // padded stride to dodge bank conflicts
  const int wid   = threadIdx.x >> 5;
  const int lane  = threadIdx.x & 31;
  const int ln    = lane & 15;
  const bool hi   = lane >= 16;
  const int c     = blockIdx.y;
  const int ebase = (blockIdx.x * WAVES + wid) * 32;
  const int e     = ebase + lane;

  unsigned int gw[5];
#pragma unroll
  for (int w = 0; w < 5; ++w) gw[w] = gcodes[((long)c * 5 + w) * NS + e];
  int st = (int)entry[(long)c * NS + e];
  float carry0 = off[(long)c * NS + ebase + ln];        // emitters ebase+0..15
  float carry1 = off[(long)c * NS + ebase + 16 + ln];   // emitters ebase+16..31

  // Constant triangular A slabs. A layout (32-bit 16x4): lane m=l&15;
  // VGPR0 holds K = (lo?0:2), VGPR1 holds K = (lo?1:3).
  v2f A[4];
#pragma unroll
  for (int q = 0; q < 4; ++q) {
    A[q].x = (ln >= 4 * q + (hi ? 2 : 0)) ? 1.0f : 0.0f;
    A[q].y = (ln >= 4 * q + (hi ? 3 : 1)) ? 1.0f : 0.0f;
  }

  int zc = 0;   // count of non-photon elements -> hist[0]
  for (int g0 = 0; g0 < 5; ++g0) {
    const unsigned int w = gw[g0];
    unsigned int pmask = 0;
#pragma unroll
    for (int i = 0; i < 16; ++i) {
      const long step = (long)c * CHUNK + g0 * 16 + i;
      const float ue = u_exp[step * NS + e];
      const int G = (int)((w >> (2 * i)) & 3u);
      const int ph = (st == 0);                 // t==1 only from S0
      const float dw = -log1pf(-ue) * invrate(st);
      out_ph[step * NS + e] = ph ? 1.0f : 0.0f;
      tile[wid][i][lane] = dw;
      pmask |= (unsigned int)ph << i;
      st = trans(st, G);
    }
    __syncthreads();

#pragma unroll
    for (int h = 0; h < 2; ++h) {
      const float carry = h ? carry1 : carry0;
      v8f d;
#pragma unroll
      for (int r = 0; r < 8; ++r) d[r] = carry;   // C = carry broadcast
#pragma unroll
      for (int q = 0; q < 4; ++q) {
        // B layout (32-bit 4x16): lane n=l&15; VGPR0 K=(lo?0:2), VGPR1 K=(lo?1:3)
        v2f b;
        b.x = tile[wid][4 * q + (hi ? 2 : 0)][h * 16 + ln];
        b.y = tile[wid][4 * q + (hi ? 3 : 1)][h * 16 + ln];
        d = __builtin_amdgcn_wmma_f32_16x16x4_f32(false, A[q], false, b,
                                                  (short)0, d, false, false);
      }
      // new carry = D[15][n] : lives in lane 16+n, VGPR 7
      const int ci = __builtin_amdgcn_ds_bpermute(4 * (16 + ln),
                                                  __float_as_int(d[7]));
      if (h) carry1 = __int_as_float(ci); else carry0 = __int_as_float(ci);
      // photon bits of this half's column (emitter ebase + h*16 + ln)
      const int pm = __builtin_amdgcn_ds_bpermute(4 * (h * 16 + ln), (int)pmask);

      int prev_ind = -1, cnt = 0;
#pragma unroll
      for (int r = 0; r < 8; ++r) {
        const int m = r + (hi ? 8 : 0);
        const float ct = d[r];
        const long idx = ((long)c * CHUNK + g0 * 16 + m) * NS + ebase + h * 16 + ln;
        out_ct[idx] = ct;
        if ((pm >> m) & 1) {
          const int ind = binIndex(ct);
          if (ind == prev_ind) ++cnt;
          else { if (cnt) atomicAdd(hist + prev_ind, cnt); prev_ind = ind; cnt = 1; }
        }
      }
      if (cnt) atomicAdd(hist + prev_ind, cnt);
      zc += 8 - __popc((unsigned)pm & (hi ? 0xFF00u : 0x00FFu));
    }
    __syncthreads();
  }
#pragma unroll
  for (int o = 16; o > 0; o >>= 1) zc += __shfl_down(zc, o, 32);
  if (lane == 0) atomicAdd(hist, zc);
}

// ---------------------------------------------------------------------------
__global__ void k_zero_hist(int* __restrict__ hist) {
  const int i = blockIdx.x * blockDim.x + threadIdx.x;
  if (i <= NBINS) hist[i] = 0;
}
__global__ void k_hist_to_float(const int* __restrict__ hist,
                                float* __restrict__ out) {
  const int i = blockIdx.x * blockDim.x + threadIdx.x;
  if (i <= NBINS) out[i] = (float)hist[i];
}

// ---------------------------------------------------------------------------
extern "C" void kernel_launch(void* const* d_in, const int* in_sizes, int n_in,
                              void* d_out, int out_size, void* d_ws, size_t ws_size,
                              hipStream_t stream) {
  const float* u_cat = (const float*)d_in[0];
  const float* u_exp = (const float*)d_in[1];

  float* out      = (float*)d_out;
  float* out_ph   = out;
  float* out_ct   = out + (size_t)STEPS * NS;
  float* out_hist = out + 2 * (size_t)STEPS * NS;

  // workspace carve-out (~24 MB total), 256B aligned slices
  char* ws = (char*)d_ws;
  size_t o = 0;
  auto take = [&](size_t bytes) -> void* {
    void* p = ws + o;
    o = (o + bytes + 255) & ~(size_t)255;
    return p;
  };
  unsigned int*  gcodes = (unsigned int*)take((size_t)NCHUNK * 5 * NS * 4); // 10.24MB
  unsigned int*  maps   = (unsigned int*)take((size_t)NCHUNK * NS * 4);     //  2.05MB
  float*         sums4  = (float*)take((size_t)NCHUNK * NS * 4 * 4);        //  8.19MB
  unsigned char* entry  = (unsigned char*)take((size_t)NCHUNK * NS);        //  0.51MB
  float*         off    = (float*)take((size_t)NCHUNK * NS * 4);            //  2.05MB
  int*           hist   = (int*)take((size_t)(NBINS + 1) * 4);              //  0.80MB
  (void)ws_size; (void)in_sizes; (void)n_in; (void)out_size;

  k_zero_hist<<<(NBINS + 256) / 256, 256, 0, stream>>>(hist);

  dim3 g1(NS / 256, NCHUNK);
  k1_codes_maps_sums<<<g1, 256, 0, stream>>>(u_cat, u_exp, gcodes, maps, sums4);

  k2_scan<<<NS / 256, 256, 0, stream>>>(maps, sums4, entry, off);

  dim3 g3(NS / (WAVES * 32), NCHUNK);
  k3_finalize<<<g3, WAVES * 32, 0, stream>>>(u_exp, gcodes, entry, off,
                                             out_ph, out_ct, hist);

  k_hist_to_float<<<(NBINS + 256) / 256, 256, 0, stream>>>(hist, out_hist);
}